// RobustCaponSolver_76965813944501
// MI455X (gfx1250) — compile-verified
//
#include <hip/hip_runtime.h>
#include <math.h>

// ---------------------------------------------------------------------------
// Robust Capon solver on MI455X (gfx1250, wave32).
// One wave = one 16x16 SPD system. Cholesky replaced by Newton-Schulz
// inversion on V_WMMA_F32_16X16X4_F32:
//   X_{k+1} = X_k (2I - Rt X_k),  X_0 = 2/(lmin+tr(Rt)) * I
// Rt is trace-normalized + shrunk + diag-loaded -> lmin >= 7.25e-3,
// tr <= ~2.45 -> 12 iterations reach f32 round-off.
// Cross-lane glue uses v_permlanex16 (half-swap) and v_readlane (broadcast)
// instead of ds_bpermute, keeping the hot loop off the LDS path.
// ---------------------------------------------------------------------------

typedef __attribute__((ext_vector_type(2))) float v2f;
typedef __attribute__((ext_vector_type(8))) float v8f;

#define GAMMA_C      1.0f
#define SIGMA_MAX_C  0.3f
#define SHRINK_C     0.1f
#define DELTA_C      1e-3f
#define EPS_C        1e-6f
#define MARGIN_EPS_C 1e-4f
#define JITTER_C     1e-6f
#define NS_ITERS     12

// lane i <- lane (i ^ 16): v_permlanex16_b32 with identity nibble selects.
__device__ __forceinline__ float xor16f(float x) {
    const int xi = __float_as_int(x);
    const int r = __builtin_amdgcn_permlanex16(xi, xi, 0x76543210, 0xFEDCBA98,
                                               /*fi=*/false, /*bound_ctrl=*/false);
    return __int_as_float(r);
}

// uniform broadcast of lane l's value: v_readlane_b32 -> SGPR operand.
__device__ __forceinline__ float bcast(float x, int l) {
    return __int_as_float(__builtin_amdgcn_readlane(__float_as_int(x), l));
}

__device__ __forceinline__ float wave_sum32(float x) {
    x += xor16f(x);                         // fold high half onto low (VALU)
#pragma unroll
    for (int off = 8; off >= 1; off >>= 1)  // within-row butterfly (cold path)
        x += __shfl_xor(x, off, 32);
    return x;
}

// D-layout (WMMA C/D): Y(m,n) at lane = n + 16*(m>=8), vgpr = m&7.
// Build A/B operand chunks for a 16x16x16 product as 4 K=4 WMMAs.
// Assumed f32 B layout (mirror of documented A layout):
//   chunk c, vgpr v holds row k = 4c+v (lanes 0-15) and k = 4c+v+2 (lanes 16-31).
// Valid as the A operand when Y is symmetric (A(m,k)=Y(k,m)).
__device__ __forceinline__ void dlayout_to_op(const float y[8], v2f out[4], bool hi) {
#pragma unroll
    for (int c = 0; c < 4; ++c) {
        float r[2];
#pragma unroll
        for (int v = 0; v < 2; ++v) {
            const int kl = 4 * c + v;      // k needed by low-half lanes
            const int kh = 4 * c + v + 2;  // k needed by high-half lanes
            float lo_val, hi_val;
            if (kl < 8)  lo_val = y[kl];             // already in this lane
            else         lo_val = xor16f(y[kl & 7]); // pull from high half
            if (kh >= 8) hi_val = y[kh & 7];         // already in this lane
            else         hi_val = xor16f(y[kh]);     // pull from low half
            r[v] = hi ? hi_val : lo_val;
        }
        v2f t = {r[0], r[1]};
        out[c] = t;
    }
}

// 16x16 (M,N) = A(16x16) x B(16x16) via four chained 16x16x4 f32 WMMAs.
__device__ __forceinline__ void mm16(const v2f A[4], const v2f B[4], float out[8]) {
    v8f acc = {0.f, 0.f, 0.f, 0.f, 0.f, 0.f, 0.f, 0.f};
#pragma unroll
    for (int c = 0; c < 4; ++c) {
        acc = __builtin_amdgcn_wmma_f32_16x16x4_f32(
            /*neg_a=*/false, A[c], /*neg_b=*/false, B[c],
            /*c_mod=*/(short)0, acc, /*reuse_a=*/false, /*reuse_b=*/false);
    }
#pragma unroll
    for (int i = 0; i < 8; ++i) out[i] = acc[i];
}

__global__ __launch_bounds__(256) void capon_ns_wmma_kernel(
    const float* __restrict__ R, const float* __restrict__ a0g,
    const float* __restrict__ sigg, float* __restrict__ wOut,
    float* __restrict__ marginOut, int P) {

    const int lane = (int)(threadIdx.x & 31u);
    const int wave = (int)(threadIdx.x >> 5u);
    const int prob = (int)blockIdx.x * 8 + wave;
    if (prob >= P) return;                 // wave-uniform: EXEC stays all-ones
    const bool hi = lane >= 16;
    const int n = lane & 15;

    const float* __restrict__ Rp = R + (size_t)prob * 256;
    float a0reg = a0g[(size_t)prob * 16 + n];
    float sreg  = sigg[(size_t)prob * 16 + n];
    sreg = fminf(fmaxf(sreg, 0.f), SIGMA_MAX_C);

    // ---- load + symmetrize R into D layout -------------------------------
    float rt[8];
#pragma unroll
    for (int v = 0; v < 8; ++v) {
        const int m = v + (hi ? 8 : 0);
        const float rmn = Rp[m * 16 + n];
        const float rnm = Rp[n * 16 + m];  // transpose read served by WGP$/L2
        rt[v] = 0.5f * (rmn + rnm);
    }

    // ---- trace normalize + shrink + diag load ----------------------------
    float tpart = 0.f;
#pragma unroll
    for (int v = 0; v < 8; ++v) {
        const bool isdiag = (lane == v) || (lane == v + 24);
        tpart += isdiag ? rt[v] : 0.f;
    }
    const float tr = wave_sum32(tpart);
    const float scale = (1.f - SHRINK_C) / fmaxf(tr, EPS_C);
    // at a diagonal position (m,m) this lane has n == m, so sreg is sigma[m]
    const float dconst = SHRINK_C / 16.f + DELTA_C + JITTER_C + GAMMA_C * sreg * sreg;
#pragma unroll
    for (int v = 0; v < 8; ++v) {
        const bool isdiag = (lane == v) || (lane == v + 24);
        rt[v] = rt[v] * scale + (isdiag ? dconst : 0.f);
    }

    // ---- Newton-Schulz init: X0 = 2/(lmin + tr(Rt)) * I ------------------
    const float ssq = wave_sum32(hi ? 0.f : sreg * sreg);
    const float trRt = (1.f - SHRINK_C) + 16.f * (SHRINK_C / 16.f + DELTA_C + JITTER_C)
                       + GAMMA_C * ssq;
    const float lminB = SHRINK_C / 16.f + DELTA_C + JITTER_C;
    const float alpha = 2.f / (trRt + lminB);

    v2f RA[4];
    dlayout_to_op(rt, RA, hi);             // Rt symmetric -> valid A operand

    float x[8];
#pragma unroll
    for (int v = 0; v < 8; ++v) {
        const bool isdiag = (lane == v) || (lane == v + 24);
        x[v] = isdiag ? alpha : 0.f;
    }

    // ---- X <- X(2I - Rt X): pure VALU + WMMA, no LDS traffic -------------
    for (int it = 0; it < NS_ITERS; ++it) {
        v2f XO[4];
        dlayout_to_op(x, XO, hi);          // X symmetric: serves as B and A
        float mM[8];
        mm16(RA, XO, mM);                  // M = Rt * X
        float nN[8];
#pragma unroll
        for (int v = 0; v < 8; ++v) {
            const bool isdiag = (lane == v) || (lane == v + 24);
            nN[v] = (isdiag ? 2.f : 0.f) - mM[v];
        }
        v2f NB[4];
        dlayout_to_op(nN, NB, hi);
        float xn[8];
        mm16(XO, NB, xn);                  // X_new = X * (2I - M)
#pragma unroll
        for (int v = 0; v < 8; ++v) x[v] = xn[v];
    }

    // ---- u = X a0 : one more WMMA with a0 replicated in every column -----
    v2f XA[4];
    dlayout_to_op(x, XA, hi);
    v2f A0B[4];
#pragma unroll
    for (int c = 0; c < 4; ++c) {
        const float l0 = bcast(a0reg, 4 * c + 0);
        const float l1 = bcast(a0reg, 4 * c + 1);
        const float h0 = bcast(a0reg, 4 * c + 2);
        const float h1 = bcast(a0reg, 4 * c + 3);
        v2f t = {hi ? h0 : l0, hi ? h1 : l1};
        A0B[c] = t;
    }
    float u[8];
    mm16(XA, A0B, u);                      // u(m) broadcast across lanes of each half

    // ---- margin, weights, softmax ----------------------------------------
    float dotp = 0.f, nrmp = 0.f;
#pragma unroll
    for (int v = 0; v < 8; ++v) {
        const float am = hi ? bcast(a0reg, v + 8) : bcast(a0reg, v);
        const float sm = hi ? bcast(sreg, v + 8) : bcast(sreg, v);
        dotp += am * u[v];
        const float t = sm * u[v];
        nrmp += t * t;
    }
    const float dot  = dotp + xor16f(dotp);
    const float nrm2 = nrmp + xor16f(nrmp);

    const float margin = dot - sqrtf(fmaxf(nrm2, 0.f));
    const bool valid = margin > MARGIN_EPS_C;
    const float denom = fmaxf(valid ? margin : 1.0f, MARGIN_EPS_C);

    float wpre[8];
#pragma unroll
    for (int v = 0; v < 8; ++v)
        wpre[v] = valid ? (u[v] / denom) : (1.0f / 16.0f);

    float mx = wpre[0];
#pragma unroll
    for (int v = 1; v < 8; ++v) mx = fmaxf(mx, wpre[v]);
    mx = fmaxf(mx, xor16f(mx));

    float ex[8];
    float s = 0.f;
#pragma unroll
    for (int v = 0; v < 8; ++v) {
        ex[v] = __expf(wpre[v] - mx);
        s += ex[v];
    }
    s += xor16f(s);
    const float inv = 1.f / s;

    // lanes 0-7 write m=0..7, lanes 16-23 write m=8..15 (each m exactly once)
    const int vsel = lane & 7;
    float wval = ex[0] * inv;
#pragma unroll
    for (int v = 1; v < 8; ++v)
        wval = (vsel == v) ? ex[v] * inv : wval;
    const bool doStore = (lane < 8) || (lane >= 16 && lane < 24);
    if (doStore) {
        const int m = vsel + (hi ? 8 : 0);
        wOut[(size_t)prob * 16 + m] = wval;
    }
    if (lane == 0) marginOut[prob] = margin;
}

extern "C" void kernel_launch(void* const* d_in, const int* in_sizes, int n_in,
                              void* d_out, int out_size, void* d_ws, size_t ws_size,
                              hipStream_t stream) {
    const float* R   = (const float*)d_in[0];
    const float* a0  = (const float*)d_in[1];
    const float* sig = (const float*)d_in[2];
    const int P = in_sizes[1] / 16;        // B*H*W independent problems
    float* w = (float*)d_out;              // [P,16]
    float* marg = w + (size_t)P * 16;      // [P]
    const int blocks = (P + 7) / 8;        // 8 waves (problems) per 256-thread block
    capon_ns_wmma_kernel<<<blocks, 256, 0, stream>>>(R, a0, sig, w, marg, P);
}